// SSI_DDI_Block_71004399337988
// MI455X (gfx1250) — compile-verified
//
#include <hip/hip_runtime.h>
#include <math.h>

typedef __attribute__((ext_vector_type(2))) float v2f;
typedef __attribute__((ext_vector_type(8))) float v8f;

#define F_IN 128
#define HC   64
#define NEG_SLOPE 0.2f

// ---------- helpers ----------

__device__ __forceinline__ float atomic_max_f32(float* addr, float val) {
  // sign-magnitude ordered-int trick; buffers are initialized to -inf
  if (val >= 0.0f)
    return __int_as_float(atomicMax((int*)addr, __float_as_int(val)));
  else
    return __uint_as_float(atomicMin((unsigned int*)addr, __float_as_uint(val)));
}

__device__ __forceinline__ float wave_sum32(float v) {
#pragma unroll
  for (int off = 16; off > 0; off >>= 1) v += __shfl_xor(v, off, 32);
  return v;
}

// sum within each 16-lane half (heads live in separate halves)
__device__ __forceinline__ float half_sum16(float v) {
#pragma unroll
  for (int off = 1; off < 16; off <<= 1) v += __shfl_xor(v, off, 32);
  return v;
}

__global__ void fill_f32(float* __restrict__ p, float v, int n) {
  int i = blockIdx.x * blockDim.x + threadIdx.x;
  if (i < n) p[i] = v;
}

// ---------- 1) xl = x @ Wl, xr = x @ Wr  (fp32 WMMA 16x16x4) ----------
// block = 256 threads = 8 waves; block handles 16 rows of x (staged in LDS).
// waves 0..3 -> 4 column tiles of xl; waves 4..7 -> 4 column tiles of xr.
__global__ __launch_bounds__(256) void gemm_xlxr_wmma(
    const float* __restrict__ x, const float* __restrict__ Wl,
    const float* __restrict__ Wr, float* __restrict__ xl,
    float* __restrict__ xr, int N) {
  __shared__ float As[16][F_IN + 4];  // +4 pad -> conflict-free column reads
  const int rowbase = blockIdx.x * 16;
  const bool fulltile = (rowbase + 16 <= N);   // block-uniform

  if (fulltile) {
    // vectorized stage: b128 global loads -> b128 LDS stores
    for (int i = threadIdx.x; i < 16 * (F_IN / 4); i += 256) {
      int r = i >> 5, k4 = (i & 31) * 4;
      float4 v = *(const float4*)&x[(size_t)(rowbase + r) * F_IN + k4];
      *(float4*)&As[r][k4] = v;
    }
  } else {
    for (int i = threadIdx.x; i < 16 * F_IN; i += 256) {
      int r = i >> 7, k = i & (F_IN - 1);
      int row = rowbase + r;
      As[r][k] = (row < N) ? x[(size_t)row * F_IN + k] : 0.0f;
    }
  }
  __syncthreads();

  const int wave = threadIdx.x >> 5;
  const int lane = threadIdx.x & 31;
  const float* __restrict__ W = (wave < 4) ? Wl : Wr;
  float* __restrict__ out     = (wave < 4) ? xl : xr;
  const int coltile = (wave & 3) * 16;
  const int mn    = lane & 15;         // M index for A frag / N index for B frag
  const int khalf = (lane >> 4) * 2;   // lanes 0-15: K+0/K+1, lanes 16-31: K+2/K+3
  const int rowoff = (lane >> 4) * 8;  // C/D: VGPR r holds M=r (lanes 0-15), M=r+8 (16-31)

  v8f acc = {};
#if __has_builtin(__builtin_amdgcn_wmma_f32_16x16x4_f32)
  for (int k0 = 0; k0 < F_IN; k0 += 4) {
    v2f a, b;
    a.x = As[mn][k0 + khalf];
    a.y = As[mn][k0 + khalf + 1];
    b.x = W[(k0 + khalf) * HC + coltile + mn];
    b.y = W[(k0 + khalf + 1) * HC + coltile + mn];
    acc = __builtin_amdgcn_wmma_f32_16x16x4_f32(
        /*neg_a=*/false, a, /*neg_b=*/false, b,
        /*c_mod=*/(short)0, acc, /*reuse_a=*/false, /*reuse_b=*/false);
  }
#else
  // layout-identical scalar fallback
#pragma unroll
  for (int r = 0; r < 8; ++r) {
    float s = 0.0f;
    for (int k = 0; k < F_IN; ++k)
      s += As[rowoff + r][k] * W[k * HC + coltile + mn];
    acc[r] = s;
  }
#endif
  if (fulltile) {
#pragma unroll
    for (int r = 0; r < 8; ++r)
      out[(size_t)(rowbase + rowoff + r) * HC + coltile + mn] = acc[r];
  } else {
#pragma unroll
    for (int r = 0; r < 8; ++r) {
      int row = rowbase + rowoff + r;
      if (row < N) out[(size_t)row * HC + coltile + mn] = acc[r];
    }
  }
}

// ---------- edge-phase common: per-wave GATv2 logits ----------
// lane l owns channels 2l, 2l+1 (b64 loads); lanes 0-15 = head 0, 16-31 = head 1.
// Returns this lane's source features and its own head's logit (in all lanes of
// the owning half after a 16-lane xor reduction).
__device__ __forceinline__ void edge_logits(
    const float* __restrict__ xl, const float* __restrict__ xr,
    const float* __restrict__ att, int s, int d, int lane,
    float2& xls, float& lg) {
  const int c = 2 * lane;
  xls       = *(const float2*)&xl[(size_t)s * HC + c];
  float2 xrd = *(const float2*)&xr[(size_t)d * HC + c];
  float a0 = xls.x + xrd.x;
  float a1 = xls.y + xrd.y;
  a0 = (a0 > 0.0f) ? a0 : NEG_SLOPE * a0;   // leaky_relu
  a1 = (a1 > 0.0f) ? a1 : NEG_SLOPE * a1;
  lg = half_sum16(a0 * att[c] + a1 * att[c + 1]);   // own-head logit
}

// ---------- 2) segment max of logits over targets ----------
__global__ __launch_bounds__(256) void edge_max_kernel(
    const int* __restrict__ src, const int* __restrict__ dst,
    const float* __restrict__ xl, const float* __restrict__ xr,
    const float* __restrict__ att, float* __restrict__ mbuf,
    long long Etot, long long Eorig) {
  long long e = (long long)blockIdx.x * 8 + (threadIdx.x >> 5);
  if (e >= Etot) return;
  int lane = threadIdx.x & 31;
  int s, d;
  if (e < Eorig) { s = src[e]; d = dst[e]; }
  else           { s = d = (int)(e - Eorig); }      // self-loop
  float2 xls; float lg;
  edge_logits(xl, xr, att, s, d, lane, xls, lg);
  int head = lane >> 4;
  if ((lane & 15) == 0)
    atomic_max_f32(&mbuf[(size_t)d * 2 + head], lg);
}

// ---------- 3) exp, softmax denominator, weighted numerator ----------
__global__ __launch_bounds__(256) void edge_accum_kernel(
    const int* __restrict__ src, const int* __restrict__ dst,
    const float* __restrict__ xl, const float* __restrict__ xr,
    const float* __restrict__ att, const float* __restrict__ mbuf,
    float* __restrict__ sbuf, float* __restrict__ num,
    long long Etot, long long Eorig) {
  long long e = (long long)blockIdx.x * 8 + (threadIdx.x >> 5);
  if (e >= Etot) return;
  int lane = threadIdx.x & 31;
  int s, d;
  if (e < Eorig) { s = src[e]; d = dst[e]; }
  else           { s = d = (int)(e - Eorig); }
  float2 xls; float lg;
  edge_logits(xl, xr, att, s, d, lane, xls, lg);
  int head = lane >> 4;
  float eh = __expf(lg - mbuf[(size_t)d * 2 + head]);   // own-head exp
  if ((lane & 15) == 0)
    atomicAdd(&sbuf[(size_t)d * 2 + head], eh);
  const int c = 2 * lane;
  atomicAdd(&num[(size_t)d * HC + c],     xls.x * eh);
  atomicAdd(&num[(size_t)d * HC + c + 1], xls.y * eh);
}

// ---------- 4) h = num/den + bias ; hr = h.w_rel ; score = h.w_root + b ----------
__global__ __launch_bounds__(256) void node_finalize_kernel(
    const float* __restrict__ num, const float* __restrict__ sbuf,
    const float* __restrict__ bias, const float* __restrict__ w_rel,
    const float* __restrict__ w_root, const float* __restrict__ b_score,
    float* __restrict__ h, float* __restrict__ hr, float* __restrict__ score,
    int N) {
  int n = blockIdx.x * 8 + (threadIdx.x >> 5);
  if (n >= N) return;
  int lane = threadIdx.x & 31;
  int head = lane >> 4;
  const int c = 2 * lane;
  float inv = 1.0f / sbuf[(size_t)n * 2 + head];   // self-loop guarantees > 0
  float2 nm = *(const float2*)&num[(size_t)n * HC + c];
  float h0 = nm.x * inv + bias[c];
  float h1 = nm.y * inv + bias[c + 1];
  *(float2*)&h[(size_t)n * HC + c] = make_float2(h0, h1);
  float pr = wave_sum32(h0 * w_rel[c]  + h1 * w_rel[c + 1]);
  float pt = wave_sum32(h0 * w_root[c] + h1 * w_root[c + 1]);
  if (lane == 0) {
    hr[n]    = pr;
    score[n] = pt + b_score[0];
  }
}

// ---------- 5) score[dst] += hr[src]  (GraphConv folded to scalar) ----------
__global__ void edge_score_kernel(const int* __restrict__ src,
                                  const int* __restrict__ dst,
                                  const float* __restrict__ hr,
                                  float* __restrict__ score, long long E) {
  long long e = (long long)blockIdx.x * blockDim.x + threadIdx.x;
  if (e < E) atomicAdd(&score[dst[e]], hr[src[e]]);
}

// ---------- 6) per-graph softmax of score ----------
__global__ void graph_max_kernel(const int* __restrict__ batch,
                                 const float* __restrict__ score,
                                 float* __restrict__ gm, int N) {
  int n = blockIdx.x * blockDim.x + threadIdx.x;
  if (n < N) atomic_max_f32(&gm[batch[n]], score[n]);
}

__global__ void graph_expsum_kernel(const int* __restrict__ batch,
                                    float* __restrict__ score,
                                    const float* __restrict__ gm,
                                    float* __restrict__ gs, int N) {
  int n = blockIdx.x * blockDim.x + threadIdx.x;
  if (n < N) {
    int b = batch[n];
    float e = __expf(score[n] - gm[b]);
    atomicAdd(&gs[b], e);
    score[n] = e;   // keep numerator for pooling pass
  }
}

// ---------- 7) weighted global add pool + node counts ----------
__global__ __launch_bounds__(256) void pool_kernel(
    const int* __restrict__ batch, const float* __restrict__ escore,
    const float* __restrict__ gs, const float* __restrict__ h,
    float* __restrict__ addp, float* __restrict__ cnt, int N) {
  int n = blockIdx.x * 8 + (threadIdx.x >> 5);
  if (n >= N) return;
  int lane = threadIdx.x & 31;
  const int c = 2 * lane;
  int b = batch[n];
  float sm = escore[n] / gs[b];
  float2 hv = *(const float2*)&h[(size_t)n * HC + c];
  atomicAdd(&addp[(size_t)b * HC + c],     hv.x * sm);
  atomicAdd(&addp[(size_t)b * HC + c + 1], hv.y * sm);
  if (lane == 0) atomicAdd(&cnt[b], 1.0f);
}

// ---------- 8) out = add + add/max(cnt,1) ----------
__global__ void final_kernel(const float* __restrict__ addp,
                             const float* __restrict__ cnt,
                             float* __restrict__ out, int total) {
  int i = blockIdx.x * blockDim.x + threadIdx.x;
  if (i < total) {
    float c = fmaxf(cnt[i / HC], 1.0f);
    out[i] = addp[i] * (1.0f + 1.0f / c);
  }
}

// ---------- launch ----------
extern "C" void kernel_launch(void* const* d_in, const int* in_sizes, int n_in,
                              void* d_out, int out_size, void* d_ws, size_t ws_size,
                              hipStream_t stream) {
  const float* x       = (const float*)d_in[0];
  const int*   ei      = (const int*)  d_in[1];
  const int*   batch   = (const int*)  d_in[2];
  const float* Wl      = (const float*)d_in[3];
  const float* Wr      = (const float*)d_in[4];
  const float* att     = (const float*)d_in[5];
  const float* bias    = (const float*)d_in[6];
  const float* w_rel   = (const float*)d_in[7];
  const float* w_root  = (const float*)d_in[8];
  const float* b_score = (const float*)d_in[9];

  const int N = in_sizes[0] / F_IN;       // 100000
  const int E = in_sizes[1] / 2;          // 1000000
  const int G = out_size / HC;            // 2048
  const int* src = ei;
  const int* dst = ei + E;

  // workspace carve (~105 MB of fp32)
  float* ws    = (float*)d_ws;
  size_t NH    = (size_t)N * HC;
  float* xl    = ws;             // N*64
  float* xr    = xl   + NH;      // N*64
  float* num   = xr   + NH;      // N*64
  float* hbuf  = num  + NH;      // N*64
  float* mbuf  = hbuf + NH;      // N*2
  float* sbuf  = mbuf + 2 * (size_t)N;   // N*2
  float* hr    = sbuf + 2 * (size_t)N;   // N
  float* score = hr   + N;               // N
  float* gm    = score + N;              // G
  float* gs    = gm   + G;               // G
  float* addp  = gs   + G;               // G*64
  float* cnt   = addp + (size_t)G * HC;  // G

  const float NEG_INF = -__builtin_huge_valf();
  auto fill = [&](float* p, float v, long long n) {
    fill_f32<<<(int)((n + 255) / 256), 256, 0, stream>>>(p, v, (int)n);
  };
  fill(num,  0.0f,    (long long)NH);
  fill(mbuf, NEG_INF, 2LL * N);
  fill(sbuf, 0.0f,    2LL * N);
  fill(gm,   NEG_INF, G);
  fill(gs,   0.0f,    G);
  fill(addp, 0.0f,    (long long)G * HC);
  fill(cnt,  0.0f,    G);

  gemm_xlxr_wmma<<<(N + 15) / 16, 256, 0, stream>>>(x, Wl, Wr, xl, xr, N);

  long long Etot = (long long)E + N;          // +self-loops
  int eblk = (int)((Etot + 7) / 8);           // 8 waves/block, 1 edge/wave
  edge_max_kernel  <<<eblk, 256, 0, stream>>>(src, dst, xl, xr, att, mbuf, Etot, E);
  edge_accum_kernel<<<eblk, 256, 0, stream>>>(src, dst, xl, xr, att, mbuf, sbuf, num, Etot, E);

  int nblk = (N + 7) / 8;
  node_finalize_kernel<<<nblk, 256, 0, stream>>>(num, sbuf, bias, w_rel, w_root,
                                                 b_score, hbuf, hr, score, N);

  edge_score_kernel<<<(int)((E + 255) / 256), 256, 0, stream>>>(src, dst, hr, score, E);
  graph_max_kernel   <<<(N + 255) / 256, 256, 0, stream>>>(batch, score, gm, N);
  graph_expsum_kernel<<<(N + 255) / 256, 256, 0, stream>>>(batch, score, gm, gs, N);
  pool_kernel<<<nblk, 256, 0, stream>>>(batch, score, gs, hbuf, addp, cnt, N);
  final_kernel<<<(G * HC + 255) / 256, 256, 0, stream>>>(addp, cnt, (float*)d_out, G * HC);
}